// GNNLayers_63471026700852
// MI455X (gfx1250) — compile-verified
//
#include <hip/hip_runtime.h>

typedef __attribute__((ext_vector_type(2))) float v2f;
typedef __attribute__((ext_vector_type(8))) float v8f;

#define N_NODES 50000
#define N_EDGES 800000
#define D       128
#define EPS     1e-5f
#define LDS_PITCH 130   // pad W rows in LDS to avoid bank aliasing between lane groups

// ---------------- degree / dinv ----------------
__global__ void deg_init_kernel(float* __restrict__ dinv) {
    int i = blockIdx.x * 256 + threadIdx.x;
    if (i < N_NODES) dinv[i] = 1.0f;   // self-loop contributes 1
}

__global__ void deg_count_kernel(const int* __restrict__ dst, float* __restrict__ dinv) {
    int e = blockIdx.x * 256 + threadIdx.x;   // grid sized exactly N_EDGES
    atomicAdd(&dinv[dst[e]], 1.0f);
}

__global__ void deg_rsqrt_kernel(float* __restrict__ dinv) {
    int i = blockIdx.x * 256 + threadIdx.x;
    if (i < N_NODES) dinv[i] = rsqrtf(dinv[i]);
}

// ---------------- h = x @ W via V_WMMA_F32_16X16X4_F32, fused self-loop epilogue ----
// Block = 256 threads = 8 waves. W (64 KB) staged into LDS once per block via
// GLOBAL_LOAD_ASYNC_TO_LDS_B128 (ASYNCcnt). Each wave owns 16 rows and computes the
// full 16x128 output stripe (8 accumulator tiles), so A is loaded exactly once.
// Epilogue writes h AND agg = h*dinv^2 + bias straight from the accumulators.
__global__ void gemm_wmma_kernel(const float* __restrict__ x,
                                 const float* __restrict__ W,
                                 const float* __restrict__ dinv,
                                 const float* __restrict__ bias,
                                 float* __restrict__ h,
                                 float* __restrict__ agg,
                                 int mtiles) {
    __shared__ float sW[D * LDS_PITCH];   // 128 rows x 130 floats = 66560 B

    // ---- async stage W (row-major 128x128) into padded LDS ----
    {
        const uint32_t lds_base = (uint32_t)(uintptr_t)sW;     // low 32 bits = LDS offset
        const uint64_t wbase    = (uint64_t)(uintptr_t)W;
        const uint32_t tid      = threadIdx.x;
        #pragma unroll
        for (int it = 0; it < 16; ++it) {                       // 4096 chunks of 16 B
            uint32_t chunk    = it * 256 + tid;
            uint32_t r        = chunk >> 5;                     // row   (32 chunks/row)
            uint32_t c        = chunk & 31;                     // chunk within row
            uint32_t lds_addr = lds_base + r * (LDS_PITCH * 4) + c * 16;
            uint32_t goff     = chunk * 16;
            asm volatile("global_load_async_to_lds_b128 %0, %1, %2"
                         :: "v"(lds_addr), "v"(goff), "s"(wbase)
                         : "memory");
        }
        asm volatile("s_wait_asynccnt 0" ::: "memory");
    }
    __syncthreads();

    const int lane   = threadIdx.x & 31;
    const int wavei  = threadIdx.x >> 5;
    const int m_tile = blockIdx.x * 8 + wavei;
    if (m_tile >= mtiles) return;                 // wave-uniform exit (after barrier)

    const int g   = lane >> 4;                    // lane-group: 0 or 1
    const int lm  = lane & 15;
    const int row = m_tile * 16 + lm;             // A: M = lane (ISA 16x4 f32 layout)

    const float* xrow = x + (size_t)row * D;

    v8f acc[8];
    #pragma unroll
    for (int n = 0; n < 8; ++n) acc[n] = {};

    for (int k0 = 0; k0 < D; k0 += 4) {
        const int ka = k0 + 2 * g;                // lane-group selects K pair {0,1}/{2,3}
        v2f a = *(const v2f*)(xrow + ka);         // A VGPR0=K=ka, VGPR1=K=ka+1
        const float* bp = sW + (size_t)ka * LDS_PITCH + lm;
        #pragma unroll
        for (int n = 0; n < 8; ++n) {
            // B 4x16: VGPR v, group g -> K = 2g+v, N = lane&15
            v2f b;
            b.x = bp[n * 16];
            b.y = bp[n * 16 + LDS_PITCH];
            acc[n] = __builtin_amdgcn_wmma_f32_16x16x4_f32(
                         false, a, false, b, (short)0, acc[n], false, false);
        }
    }

    // per-output-row dinv^2 (rows m_tile*16 + 8g + r, broadcast loads)
    float dv2[8];
    #pragma unroll
    for (int r = 0; r < 8; ++r) {
        float dv = dinv[m_tile * 16 + 8 * g + r];
        dv2[r] = dv * dv;
    }

    // C/D 16x16 f32: VGPR r -> M = r + 8*g, N = lane&15
    // Fused epilogue: h = acc ; agg = acc*dinv^2 + bias (self-loop term)
    #pragma unroll
    for (int n = 0; n < 8; ++n) {
        const size_t base = (size_t)(m_tile * 16 + 8 * g) * D + n * 16 + lm;
        float* hp = h   + base;
        float* ap = agg + base;
        float  bc = bias[n * 16 + lm];
        #pragma unroll
        for (int r = 0; r < 8; ++r) {
            float v = acc[n][r];
            hp[(size_t)r * D] = v;
            ap[(size_t)r * D] = v * dv2[r] + bc;
        }
    }
}

// ---------------- edge scatter: agg[dst] += h[src] * dinv[src]*dinv[dst] ----------------
__global__ void scatter_kernel(const float* __restrict__ h,
                               const int* __restrict__ src,
                               const int* __restrict__ dst,
                               const float* __restrict__ dinv,
                               float* __restrict__ agg) {
    int idx = blockIdx.x * 256 + threadIdx.x;   // N_EDGES*32 threads, exact
    int e = idx >> 5;
    int q = (idx & 31) * 4;
    int s = src[e];
    int t = dst[e];
    float norm = dinv[s] * dinv[t];
    float4 v = *(const float4*)(h + (size_t)s * D + q);   // L2-resident gather
    float* ap = agg + (size_t)t * D + q;
    atomicAdd(ap + 0, v.x * norm);
    atomicAdd(ap + 1, v.y * norm);
    atomicAdd(ap + 2, v.z * norm);
    atomicAdd(ap + 3, v.w * norm);
}

// ---------------- LayerNorm + ReLU + residual (one node per wave32) ----------------
__global__ void ln_relu_res_kernel(const float* __restrict__ agg,
                                   const float* __restrict__ gamma,
                                   const float* __restrict__ beta,
                                   const float* __restrict__ xprev,
                                   float* __restrict__ out,
                                   int residual) {
    const int lane = threadIdx.x & 31;
    const int wave = threadIdx.x >> 5;
    const int node = blockIdx.x * 8 + wave;     // 6250*8 = 50000, exact
    const int q = lane * 4;

    float4 v = *(const float4*)(agg + (size_t)node * D + q);

    float sum = v.x + v.y + v.z + v.w;
    #pragma unroll
    for (int off = 16; off > 0; off >>= 1) sum += __shfl_xor(sum, off, 32);
    float mu = sum * (1.0f / D);

    float dx = v.x - mu, dy = v.y - mu, dz = v.z - mu, dw = v.w - mu;
    float ss = dx * dx + dy * dy + dz * dz + dw * dw;
    #pragma unroll
    for (int off = 16; off > 0; off >>= 1) ss += __shfl_xor(ss, off, 32);
    float rs = rsqrtf(ss * (1.0f / D) + EPS);

    float4 gm = *(const float4*)(gamma + q);
    float4 bt = *(const float4*)(beta + q);
    float4 o;
    o.x = fmaxf(dx * rs * gm.x + bt.x, 0.0f);
    o.y = fmaxf(dy * rs * gm.y + bt.y, 0.0f);
    o.z = fmaxf(dz * rs * gm.z + bt.z, 0.0f);
    o.w = fmaxf(dw * rs * gm.w + bt.w, 0.0f);

    if (residual) {
        float4 p = *(const float4*)(xprev + (size_t)node * D + q);  // in-place safe
        o.x += p.x; o.y += p.y; o.z += p.z; o.w += p.w;
    }
    *(float4*)(out + (size_t)node * D + q) = o;
}

// ---------------- driver ----------------
extern "C" void kernel_launch(void* const* d_in, const int* in_sizes, int n_in,
                              void* d_out, int out_size, void* d_ws, size_t ws_size,
                              hipStream_t stream) {
    const float* x      = (const float*)d_in[0];
    const int*   edge   = (const int*)d_in[1];
    const float* Ws     = (const float*)d_in[2];
    const float* bs     = (const float*)d_in[3];
    const float* gammas = (const float*)d_in[4];
    const float* betas  = (const float*)d_in[5];

    const int E = in_sizes[1] / 2;            // 800000
    const int* src = edge;
    const int* dst = edge + E;

    float* out  = (float*)d_out;
    float* ws   = (float*)d_ws;
    float* dinv = ws;                          // 50000 floats (padded to 50048)
    float* hbuf = ws + 50048;                  // 50000*128 floats
    float* aggb = hbuf + (size_t)N_NODES * D;  // 50000*128 floats

    const int mtiles = N_NODES / 16;           // 3125

    // degrees -> dinv
    deg_init_kernel <<<(N_NODES + 255) / 256, 256, 0, stream>>>(dinv);
    deg_count_kernel<<<N_EDGES / 256,        256, 0, stream>>>(dst, dinv);
    deg_rsqrt_kernel<<<(N_NODES + 255) / 256, 256, 0, stream>>>(dinv);

    for (int l = 0; l < 3; ++l) {
        const float* xin = (l == 0) ? x : out;
        gemm_wmma_kernel  <<<(mtiles + 7) / 8,   256, 0, stream>>>(xin, Ws + (size_t)l * D * D,
                                                                   dinv, bs + l * D, hbuf, aggb, mtiles);
        scatter_kernel    <<<N_EDGES * 32 / 256, 256, 0, stream>>>(hbuf, src, dst, dinv, aggb);
        ln_relu_res_kernel<<<N_NODES / 8,        256, 0, stream>>>(aggb, gammas + l * D, betas + l * D,
                                                                   out, out, l > 0);
    }
}